// PieceMaxPool_64304250355848
// MI455X (gfx1250) — compile-verified
//
#include <hip/hip_runtime.h>
#include <math.h>

// Problem constants (match reference: B=256, C=768, L=256, P=3, MINUS=-100)
#define B_     256
#define C_     768
#define L_     256
#define P_     3
#define MINUSV (-100.0f)
#define CCHUNK 64          // channels per block (8 per wave * 8 waves)
#define NWAVES 8
#define NBUF   4           // LDS buffers per wave (depth-2 async pipeline)

// One async b128 pair copies a full 1KB row (L_*4 bytes) into LDS:
//   instr 0: bytes [0,512)  (lane*16 each), instr 1: bytes [512,1024) via offset:512
// INST_OFFSET applies to BOTH the LDS destination and the global address (ISA 08 §4.4),
// so a single address pair drives both halves. NT hint: x has zero reuse.
#define ASYNC_ROW(ldsaddr, gptr)                                                   \
  asm volatile("global_load_async_to_lds_b128 %0, %1, off th:TH_LOAD_NT\n\t"      \
               "global_load_async_to_lds_b128 %0, %1, off offset:512 th:TH_LOAD_NT"\
               :: "v"(ldsaddr), "v"(gptr) : "memory")

__global__ __launch_bounds__(256) void
PieceMaxPool_kernel(const float* __restrict__ x,
                    const int*   __restrict__ mask,
                    float*       __restrict__ out)
{
  // 8 waves * 4 buffers * 256 floats = 32 KB (WGP has 320 KB)
  __shared__ float ldsX[NWAVES][NBUF][L_];

  const int tid  = threadIdx.x;
  const int lane = tid & 31;
  const int w    = tid >> 5;

  const int b     = blockIdx.x / (C_ / CCHUNK);
  const int cbase = (blockIdx.x % (C_ / CCHUNK)) * CCHUNK + w * 8;

  // ---- Per-lane piece bias, computed once per wave (reused for 8 channels) ----
  // Lane handles l = lane*4+j (j=0..3) and l = 128+lane*4+j  -> 8 positions.
  const int4* mrow = (const int4*)(mask + (size_t)b * L_);
  const int4  ma   = mrow[lane];        // l = lane*4 .. lane*4+3
  const int4  mb   = mrow[lane + 32];   // l = 128+lane*4 ..
  int mv[8] = {ma.x, ma.y, ma.z, ma.w, mb.x, mb.y, mb.z, mb.w};

  float bias[P_][8];
#pragma unroll
  for (int p = 0; p < P_; ++p)
#pragma unroll
    for (int j = 0; j < 8; ++j)
      bias[p][j] = (mv[j] == p + 1) ? 0.0f : MINUSV;

  // ---- Async double... quad-buffered row pipeline ----
  const float* rowp = x + ((size_t)b * C_ + (size_t)cbase) * L_ + lane * 4; // lane*16B
  const unsigned ldsbase =
      (unsigned)(uintptr_t)(&ldsX[w][0][0]) + (unsigned)lane * 16u;

  // Prologue: prefetch rows 0 and 1 (ASYNCcnt = 4)
  ASYNC_ROW(ldsbase + 0u * (L_ * 4u), rowp + 0 * L_);
  ASYNC_ROW(ldsbase + 1u * (L_ * 4u), rowp + 1 * L_);

#pragma unroll
  for (int i = 0; i < 8; ++i) {
    if (i + 2 < 8) {
      const unsigned la = ldsbase + (unsigned)((i + 2) & (NBUF - 1)) * (L_ * 4u);
      ASYNC_ROW(la, rowp + (size_t)(i + 2) * L_);
    }
    // Row i complete when ASYNCcnt <= 2 * (#newer rows in flight); loads retire in order.
    if (i < 6)       asm volatile("s_wait_asynccnt 4" ::: "memory");
    else if (i == 6) asm volatile("s_wait_asynccnt 2" ::: "memory");
    else             asm volatile("s_wait_asynccnt 0" ::: "memory");

    // Conflict-free b128 LDS reads: lane slice = floats [lane*4, +4) and [128+lane*4, +4)
    const float4 xa = *(const float4*)&ldsX[w][i & (NBUF - 1)][lane * 4];
    const float4 xb = *(const float4*)&ldsX[w][i & (NBUF - 1)][128 + lane * 4];
    const float xv[8] = {xa.x, xa.y, xa.z, xa.w, xb.x, xb.y, xb.z, xb.w};

    float m0 = -INFINITY, m1 = -INFINITY, m2 = -INFINITY;
#pragma unroll
    for (int j = 0; j < 8; ++j) {
      m0 = fmaxf(m0, xv[j] + bias[0][j]);
      m1 = fmaxf(m1, xv[j] + bias[1][j]);
      m2 = fmaxf(m2, xv[j] + bias[2][j]);
    }

    // wave32 xor-shuffle max tree (5 steps)
#pragma unroll
    for (int off = 16; off > 0; off >>= 1) {
      m0 = fmaxf(m0, __shfl_xor(m0, off, 32));
      m1 = fmaxf(m1, __shfl_xor(m1, off, 32));
      m2 = fmaxf(m2, __shfl_xor(m2, off, 32));
    }

    if (lane == 0) {
      float* o = out + (size_t)b * (P_ * C_) + (size_t)(cbase + i);
      o[0]        = m0;   // piece 0
      o[C_]       = m1;   // piece 1
      o[2 * C_]   = m2;   // piece 2
    }
  }
}

extern "C" void kernel_launch(void* const* d_in, const int* in_sizes, int n_in,
                              void* d_out, int out_size, void* d_ws, size_t ws_size,
                              hipStream_t stream) {
  const float* x    = (const float*)d_in[0];
  const int*   mask = (const int*)d_in[1];   // int inputs arrive as int32
  float*       out  = (float*)d_out;

  const int grid = B_ * (C_ / CCHUNK);       // 256 * 12 = 3072 workgroups
  PieceMaxPool_kernel<<<grid, NWAVES * 32, 0, stream>>>(x, mask, out);
}